// InpaintingBlock_10376640987473
// MI455X (gfx1250) — compile-verified
//
#include <hip/hip_runtime.h>

typedef float v2f __attribute__((ext_vector_type(2)));
typedef float v8f __attribute__((ext_vector_type(8)));

#define IMGH 1024
#define IMGW 1024
#define TILE 128
#define P    180        // LDS buffer dim: 178 data cells + 1 zero border each side
#define NSUB 11         // 11x11 subtiles of 16x16 cover the 176-wide update region
#define NST  (NSUB*NSUB) // 121

// One workgroup = one 128x128 output tile of one (batch,channel) image.
// All 20 diffusion steps fused in LDS (temporal blocking, halo=25).
// Stencil per 16x16 subtile runs on the matrix pipe as
//   D = A_band(16x20) x X(20x16)  +  X(16x20) x B_band(20x16)
// via 10 chained V_WMMA_F32_16X16X4_F32. All 15 LDS operand loads per
// subtile are issued before the WMMA chain so dscnt waits stay partial.
__global__ __launch_bounds__(1024) void inpaint_fused_kernel(
    const float* __restrict__ x, const float* __restrict__ wgt,
    float* __restrict__ out, const int* __restrict__ tsp)
{
    __shared__ float buf[P * P];   // 129,600 B

    const int tid  = threadIdx.x;
    const int lane = tid & 31;
    const int wave = tid >> 5;     // 0..31
    const int hi   = lane >> 4;    // half-wave select (0/1)
    const int nn   = lane & 15;    // row/col index within half

    const int blk    = blockIdx.x;   // 0..1023
    const int img    = blk >> 6;     // 0..15  == b*2 + ch
    const int tr     = (blk >> 3) & 7;
    const int tc     = blk & 7;
    const int ch     = img & 1;
    const int base_r = tr * TILE;
    const int base_c = tc * TILE;
    const float* __restrict__ xin  = x   + (size_t)img * IMGH * IMGW;
    float*       __restrict__ xout = out + (size_t)img * IMGH * IMGW;

    int steps = tsp[0];
    if (steps < 0)  steps = 0;
    if (steps > 20) steps = 20;    // halo sized for 20 fused steps

    // 5-point stencil weights for this channel (3x3 kernel, OIHW)
    const float w_top = wgt[ch * 9 + 1];
    const float w_lft = wgt[ch * 9 + 3];
    const float w_ctr = wgt[ch * 9 + 4];
    const float w_rgt = wgt[ch * 9 + 5];
    const float w_bot = wgt[ch * 9 + 7];

    // ---------------- load phase: fill P x P buffer ----------------
    // buffer (i,j), data region i,j in [1,179): image g = base + i - 26,
    // reflect-mirrored into [0,1024). Border ring zeroed (read by zero-weight
    // WMMA rows/cols; must not be NaN).
    for (int idx = tid; idx < P * P; idx += 1024) {
        int i = idx / P, j = idx - i * P;
        float v = 0.0f;
        if (i >= 1 && i < P - 1 && j >= 1 && j < P - 1) {
            int gr = base_r + i - 26;
            int gc = base_c + j - 26;
            gr = (gr < 0) ? -gr : ((gr >= IMGH) ? (2 * IMGH - 2 - gr) : gr);
            gc = (gc < 0) ? -gc : ((gc >= IMGW) ? (2 * IMGW - 2 - gc) : gc);
            v = xin[gr * IMGW + gc];
        }
        buf[idx] = v;
    }
    __syncthreads();

    // ---------------- constant banded matrix operands ----------------
    // Window index q in 0..19 maps to input offset q-2 in -2..17.
    // Vertical A'[m][q]: w_top at q==m+1, w_ctr at q==m+2, w_bot at q==m+3.
    // Horizontal B'[q][n]: w_lft at q==n+1, w_rgt at q==n+3 (center in vertical).
    v2f aV[5], bH[5];
#pragma unroll
    for (int k = 0; k < 5; ++k) {
        const int q0 = 4 * k + 2 * hi;
        const int q1 = q0 + 1;
        aV[k].x = (q0 == nn + 1) ? w_top : (q0 == nn + 2) ? w_ctr : (q0 == nn + 3) ? w_bot : 0.0f;
        aV[k].y = (q1 == nn + 1) ? w_top : (q1 == nn + 2) ? w_ctr : (q1 == nn + 3) ? w_bot : 0.0f;
        bH[k].x = (q0 == nn + 1) ? w_lft : (q0 == nn + 3) ? w_rgt : 0.0f;
        bH[k].y = (q1 == nn + 1) ? w_lft : (q1 == nn + 3) ? w_rgt : 0.0f;
    }

    // ---------------- subtile assignment (4 per wave) ----------------
    int sor[4], soc[4];
#pragma unroll
    for (int s = 0; s < 4; ++s) {
        int t  = wave + 32 * s;
        int tt = (t < NST) ? t : 0;
        sor[s] = 2 + 16 * (tt / NSUB);
        soc[s] = 2 + 16 * (tt % NSUB);
    }
    const bool valid3 = (wave + 96) < NST;   // s=0..2 always valid (max id 95)

    // ---------------- mask bits: 1 = unknown (x==0) -> updated ----------------
    unsigned mbits = 0;
#pragma unroll
    for (int s = 0; s < 4; ++s) {
        if (s < 3 || valid3) {
#pragma unroll
            for (int r = 0; r < 8; ++r) {
                float v = buf[(sor[s] + r + 8 * hi) * P + soc[s] + nn];
                if (v == 0.0f) mbits |= (1u << (8 * s + r));
            }
        }
    }

    // D = Tv x X + X x Th on one 16x16 subtile (origin orr,occ in buf coords).
    // Issue all 15 LDS loads (10 b32 + effectively 5 b64) first, then run the
    // 10-deep WMMA accumulator chain.
    auto computeSub = [&](int orr, int occ) -> v8f {
        v2f bV[5], aH[5];
#pragma unroll
        for (int k = 0; k < 5; ++k) {          // vertical B = X rows (2 x b32)
            const int r0 = 4 * k - 2 + 2 * hi;
            bV[k].x = buf[(orr + r0)     * P + occ + nn];
            bV[k].y = buf[(orr + r0 + 1) * P + occ + nn];
        }
#pragma unroll
        for (int k = 0; k < 5; ++k) {          // horizontal A = X col pairs (b64)
            const int c0 = 4 * k - 2 + 2 * hi;
            aH[k].x = buf[(orr + nn) * P + occ + c0];
            aH[k].y = buf[(orr + nn) * P + occ + c0 + 1];
        }
        v8f d = (v8f)0.0f;
#pragma unroll
        for (int k = 0; k < 5; ++k)
            d = __builtin_amdgcn_wmma_f32_16x16x4_f32(false, aV[k], false, bV[k],
                                                      (short)0, d, false, false);
#pragma unroll
        for (int k = 0; k < 5; ++k)
            d = __builtin_amdgcn_wmma_f32_16x16x4_f32(false, aH[k], false, bH[k],
                                                      (short)0, d, false, false);
        return d;
    };

    auto storeSub = [&](int orr, int occ, v8f d, int s) {
#pragma unroll
        for (int r = 0; r < 8; ++r) {
            if ((mbits >> (8 * s + r)) & 1u)   // only unknown pixels are written
                buf[(orr + r + 8 * hi) * P + occ + nn] = d[r];
        }
    };

    // ---------------- fused time stepping ----------------
    for (int t = 0; t < steps; ++t) {
        v8f d0 = computeSub(sor[0], soc[0]);
        v8f d1 = computeSub(sor[1], soc[1]);
        v8f d2 = computeSub(sor[2], soc[2]);
        v8f d3 = d0;
        if (valid3) d3 = computeSub(sor[3], soc[3]);
        __syncthreads();                        // all reads done before any write
        storeSub(sor[0], soc[0], d0, 0);
        storeSub(sor[1], soc[1], d1, 1);
        storeSub(sor[2], soc[2], d2, 2);
        if (valid3) storeSub(sor[3], soc[3], d3, 3);
        __syncthreads();                        // all writes done before next read
    }

    // ---------------- output: central 128x128 ----------------
    for (int idx = tid; idx < TILE * TILE; idx += 1024) {
        int u = idx >> 7, v = idx & 127;
        xout[(base_r + u) * IMGW + base_c + v] = buf[(u + 26) * P + (v + 26)];
    }
}

extern "C" void kernel_launch(void* const* d_in, const int* in_sizes, int n_in,
                              void* d_out, int out_size, void* d_ws, size_t ws_size,
                              hipStream_t stream)
{
    const float* x   = (const float*)d_in[0];
    const float* w   = (const float*)d_in[1];
    const int*   ts  = (const int*)d_in[2];
    float*       out = (float*)d_out;
    // 8 batches * 2 channels * 8x8 tiles = 1024 workgroups, 1024 threads each
    inpaint_fused_kernel<<<dim3(1024), dim3(1024), 0, stream>>>(x, w, out, ts);
}